// SSIM_3350074491442
// MI455X (gfx1250) — compile-verified
//
#include <hip/hip_runtime.h>
#include <cmath>

// ---------------------------------------------------------------------------
// SSIM fused kernel for MI455X (gfx1250, wave32).
// Separable 11x11 Gaussian conv expressed as band-matrix GEMMs on
// V_WMMA_F32_16X16X4_F32 (fp32 in/out, numerically equivalent to reference).
// ---------------------------------------------------------------------------

typedef float v2f __attribute__((ext_vector_type(2)));
typedef float v8f __attribute__((ext_vector_type(8)));

#define WSZ      11
#define PADR     5
#define TILE     32
#define IN_ROWS  48     // 42 valid rows + padding up to 3x16 row chunks
#define IN_PITCH 45     // 44 valid cols (c0=16 chunk reads up to col 43); odd pitch -> conflict-free column walks
#define R_ROWS   48
#define R_PITCH  44     // 8*44 mod 64 = 32 and 2*44 mod 64 = 24 -> lane-half accesses hit disjoint banks
#define KSTEPS   7      // ceil(26/4): K padded 26 -> 28, zero taps in the band
#define NPLANES  5      // x, y, x^2, y^2, xy
#define IMGW     512
#define IMGH     512

// D = A(16x4) * B(4x16) + C, fp32. 8-arg VOP3P form per probe doc.
__device__ __forceinline__ v8f wmma4(v2f a, v2f b, v8f c) {
  return __builtin_amdgcn_wmma_f32_16x16x4_f32(
      /*neg_a=*/false, a, /*neg_b=*/false, b,
      /*c_mod=*/(short)0, c, /*reuse_a=*/false, /*reuse_b=*/false);
}

// Order-preserving float <-> uint map for min/max atomics.
__device__ __forceinline__ unsigned f2ord(float f) {
  unsigned u = __float_as_uint(f);
  return (u & 0x80000000u) ? ~u : (u | 0x80000000u);
}
__device__ __forceinline__ float ord2f(unsigned u) {
  u = (u & 0x80000000u) ? (u & 0x7fffffffu) : ~u;
  return __uint_as_float(u);
}

// Gaussian band value g[idx], zero outside [0,10]. sg points at LDS table.
__device__ __forceinline__ float gband(const float* sg, int idx) {
  int ci = idx < 0 ? 0 : (idx > 10 ? 10 : idx);
  float v = sg[ci];
  return (idx == ci) ? v : 0.0f;
}

// --------------------------- reduction kernels -----------------------------

__global__ void ssim_init(unsigned* ws) {
  ws[0] = 0u;           // running ord-max
  ws[1] = 0xFFFFFFFFu;  // running ord-min
}

__global__ __launch_bounds__(256) void ssim_minmax(const float* __restrict__ x,
                                                   unsigned n, unsigned* ws) {
  unsigned i = blockIdx.x * blockDim.x + threadIdx.x;
  const unsigned stride = gridDim.x * blockDim.x;
  float mn = INFINITY, mx = -INFINITY;
  for (; i < n; i += stride) {
    float v = x[i];
    mn = fminf(mn, v);
    mx = fmaxf(mx, v);
  }
  // wave32 butterfly reduce
  #pragma unroll
  for (int off = 16; off; off >>= 1) {
    mn = fminf(mn, __shfl_xor(mn, off, 32));
    mx = fmaxf(mx, __shfl_xor(mx, off, 32));
  }
  __shared__ float smn[8], smx[8];
  const int wv = threadIdx.x >> 5;
  if ((threadIdx.x & 31) == 0) { smn[wv] = mn; smx[wv] = mx; }
  __syncthreads();
  if (threadIdx.x == 0) {
    const int nw = blockDim.x >> 5;
    for (int w = 1; w < nw; ++w) { mn = fminf(mn, smn[w]); mx = fmaxf(mx, smx[w]); }
    atomicMax(&ws[0], f2ord(mx));
    atomicMin(&ws[1], f2ord(mn));
  }
}

// ------------------------------ main kernel --------------------------------
// One 128-thread block (4 waves) per 32x32 output tile of one batch image.

__global__ __launch_bounds__(128) void ssim_main(const float* __restrict__ img1,
                                                 const float* __restrict__ img2,
                                                 float* __restrict__ out,
                                                 const unsigned* __restrict__ ws) {
  __shared__ float s_in1[IN_ROWS * IN_PITCH];
  __shared__ float s_in2[IN_ROWS * IN_PITCH];
  __shared__ float s_R[NPLANES][R_ROWS * R_PITCH];
  __shared__ float s_g[WSZ];

  const int tid  = threadIdx.x;
  const int wave = tid >> 5;
  const int lane = tid & 31;
  const int half = lane >> 4;  // lane half: K/M split per WMMA VGPR layout
  const int lm   = lane & 15;  // M (A rows) or N (B/C/D cols) index

  // --- Gaussian taps (matches torch: center at ws/2 = 5.5, sigma = 1.5) ---
  if (tid == 0) {
    float tmp[WSZ];
    float sum = 0.0f;
    #pragma unroll
    for (int i = 0; i < WSZ; ++i) {
      float x = (float)i - 5.5f;
      tmp[i] = expf(-(x * x) / 4.5f);  // 2*sigma^2 = 4.5
      sum += tmp[i];
    }
    #pragma unroll
    for (int i = 0; i < WSZ; ++i) s_g[i] = tmp[i] / sum;
  }

  // --- cooperative tile load with zero padding (pad = 5, SAME conv) ---
  const int tR = blockIdx.y * TILE - PADR;
  const int tC = blockIdx.x * TILE - PADR;
  const float* __restrict__ b1 = img1 + (size_t)blockIdx.z * IMGW * IMGH;
  const float* __restrict__ b2 = img2 + (size_t)blockIdx.z * IMGW * IMGH;
  for (int idx = tid; idx < IN_ROWS * IN_PITCH; idx += blockDim.x) {
    const int r = idx / IN_PITCH;
    const int c = idx - r * IN_PITCH;
    const int gr = tR + r, gc = tC + c;
    float v1 = 0.0f, v2 = 0.0f;
    if (gr >= 0 && gr < IMGH && gc >= 0 && gc < IMGW) {
      const int o = gr * IMGW + gc;
      v1 = b1[o];
      v2 = b2[o];
    }
    s_in1[idx] = v1;
    s_in2[idx] = v2;
  }
  __syncthreads();

  // --- band operand chunks: band[kc].{x,y} = g[(4kc + 2*half + {0,1}) - lm]
  // Serves as B (horizontal pass: W[j][c]=g[j-c]) and A (vertical: Wv[m][j]=g[j-m]).
  v2f band[KSTEPS];
  #pragma unroll
  for (int kc = 0; kc < KSTEPS; ++kc) {
    const int j0 = 4 * kc + 2 * half;
    band[kc].x = gband(s_g, j0 - lm);
    band[kc].y = gband(s_g, j0 + 1 - lm);
  }

  // ---------------- horizontal pass: R_p = row-conv of plane p --------------
  // 6 chunks: row base r0 in {0,16,32} (42 valid rows + safe garbage), col base c0 in {0,16}.
  for (int chunk = wave; chunk < 6; chunk += 4) {
    const int r0 = (chunk >> 1) * 16;
    const int c0 = (chunk & 1) * 16;
    v8f acc[NPLANES] = {};
    #pragma unroll
    for (int kc = 0; kc < KSTEPS; ++kc) {
      const int col = c0 + 4 * kc + 2 * half;
      const int row = r0 + lm;
      const int o = row * IN_PITCH + col;
      const float x0 = s_in1[o], x1 = s_in1[o + 1];
      const float y0 = s_in2[o], y1 = s_in2[o + 1];
      const v2f b = band[kc];
      v2f a;
      a.x = x0;      a.y = x1;      acc[0] = wmma4(a, b, acc[0]);
      a.x = y0;      a.y = y1;      acc[1] = wmma4(a, b, acc[1]);
      a.x = x0 * x0; a.y = x1 * x1; acc[2] = wmma4(a, b, acc[2]);
      a.x = y0 * y0; a.y = y1 * y1; acc[3] = wmma4(a, b, acc[3]);
      a.x = x0 * y0; a.y = x1 * y1; acc[4] = wmma4(a, b, acc[4]);
    }
    // scatter D (16x16 f32 layout: lane half -> M+8, VGPR v -> M row) into R planes
    #pragma unroll
    for (int p = 0; p < NPLANES; ++p) {
      #pragma unroll
      for (int v = 0; v < 8; ++v) {
        const int row = r0 + half * 8 + v;
        s_R[p][row * R_PITCH + c0 + lm] = acc[p][v];
      }
    }
  }
  __syncthreads();

  // ---------------- vertical pass + SSIM math ------------------------------
  // quadrant per wave: r0/c0 in {0,16}
  const int r0 = (wave >> 1) * 16;
  const int c0 = (wave & 1) * 16;
  v8f acc[NPLANES] = {};
  #pragma unroll
  for (int kc = 0; kc < KSTEPS; ++kc) {
    const int rr = r0 + 4 * kc + 2 * half;
    const int cc = c0 + lm;
    const v2f a = band[kc];
    #pragma unroll
    for (int p = 0; p < NPLANES; ++p) {
      v2f b;
      b.x = s_R[p][rr * R_PITCH + cc];
      b.y = s_R[p][(rr + 1) * R_PITCH + cc];
      acc[p] = wmma4(a, b, acc[p]);
    }
  }

  // dynamic range constants (uniform scalar loads)
  float L = ord2f(ws[0]) - ord2f(ws[1]);
  if (L == 0.0f) L = 5.0f;
  const float C1 = (0.01f * L) * (0.01f * L);
  const float C2 = (0.03f * L) * (0.03f * L);

  float* __restrict__ op = out + (size_t)blockIdx.z * IMGW * IMGH;
  const int gr0 = blockIdx.y * TILE + r0 + half * 8;
  const int gc  = blockIdx.x * TILE + c0 + lm;
  #pragma unroll
  for (int v = 0; v < 8; ++v) {
    const float mu1 = acc[0][v], mu2 = acc[1][v];
    const float e11 = acc[2][v], e22 = acc[3][v], e12 = acc[4][v];
    const float mu1s = mu1 * mu1, mu2s = mu2 * mu2, mu12 = mu1 * mu2;
    const float s1 = e11 - mu1s, s2 = e22 - mu2s, s12 = e12 - mu12;
    const float num = (2.0f * mu12 + C1) * (2.0f * s12 + C2);
    const float den = (mu1s + mu2s + C1) * (s1 + s2 + C2);
    op[(gr0 + v) * IMGW + gc] = num / den;
  }
}

// ------------------------------- launcher ----------------------------------

extern "C" void kernel_launch(void* const* d_in, const int* in_sizes, int n_in,
                              void* d_out, int out_size, void* d_ws, size_t ws_size,
                              hipStream_t stream) {
  const float* img1 = (const float*)d_in[0];
  const float* img2 = (const float*)d_in[1];
  float* out = (float*)d_out;
  unsigned* ws = (unsigned*)d_ws;

  const unsigned n = (unsigned)in_sizes[0];          // 32*1*512*512
  const int batch = (int)(n / (IMGW * IMGH));

  ssim_init<<<1, 1, 0, stream>>>(ws);
  ssim_minmax<<<2048, 256, 0, stream>>>(img1, n, ws);

  dim3 grid(IMGW / TILE, IMGH / TILE, batch);
  ssim_main<<<grid, 128, 0, stream>>>(img1, img2, out, ws);
}